// NaiveFusionGNN_24481313587803
// MI455X (gfx1250) — compile-verified
//
#include <hip/hip_runtime.h>

#define IN_F 128

typedef float v2f __attribute__((ext_vector_type(2)));
typedef float v4f __attribute__((ext_vector_type(4)));
typedef float v8f __attribute__((ext_vector_type(8)));

// ---------------- degree kernels ----------------
__global__ void k_deg_init(float* __restrict__ deg, int n) {
  int i = blockIdx.x * blockDim.x + threadIdx.x;
  if (i < n) deg[i] = 1.0f;   // self-loop counted
}

__global__ void k_deg_count(const int* __restrict__ dst, float* __restrict__ deg, int e) {
  int i = blockIdx.x * blockDim.x + threadIdx.x;
  if (i < e) atomicAdd(&deg[dst[i]], 1.0f);
}

__global__ void k_deg_fin(const float* __restrict__ deg, float* __restrict__ dis,
                          float* __restrict__ di, int n) {
  int i = blockIdx.x * blockDim.x + threadIdx.x;
  if (i < n) {
    float d = deg[i];
    dis[i] = 1.0f / sqrtf(d);
    di[i]  = 1.0f / d;
  }
}

__global__ void k_zero4(v4f* __restrict__ p, long n4) {
  long i = (long)blockIdx.x * blockDim.x + threadIdx.x;
  if (i < n4) p[i] = v4f{0.f, 0.f, 0.f, 0.f};
}

// ---------------- WMMA fp32 GEMM: Y[N x FOUT] = X[N x 128] @ W[128 x FOUT] ----------------
// MODE 0: Y = acc             (bias unused)
// MODE 1: Y = relu(acc + b)
// MODE 2: Y = 0.5*(acc + b)
template <int FOUT, int MODE>
__global__ __launch_bounds__(256) void k_gemm_wmma(
    const float* __restrict__ X, const float* __restrict__ W,
    const float* __restrict__ bias, float* __restrict__ Y, int nrows) {
  constexpr int LSTR = IN_F + 2;               // padded k-stride (keeps float2 8B aligned, k even)
  __shared__ float Wl[FOUT * LSTR];            // W transposed: Wl[n*LSTR + k]

  const int tid = threadIdx.x;
  // stage W (row-major [k][FOUT] in memory) transposed into LDS
  for (int i = tid; i < IN_F * FOUT; i += 256) {
    int k = i / FOUT, n = i % FOUT;
    Wl[n * LSTR + k] = W[i];
  }
  __syncthreads();

  const int lane = tid & 31;
  const int wave = tid >> 5;                   // 0..7
  const int rowBase = blockIdx.x * 128 + wave * 16;
  const int m = lane & 15;                     // A row / B col / D col within tile
  const int ksel = (lane >> 4) << 1;           // half-wave K offset: 0 or 2

  int arow = rowBase + m;
  if (arow >= nrows) arow = nrows - 1;         // clamp: keep EXEC all-1s for WMMA
  const float* Xrow = X + (size_t)arow * IN_F;

  constexpr int NT = FOUT / 16;
  v8f acc[NT];
  v8f zero8 = {0.f, 0.f, 0.f, 0.f, 0.f, 0.f, 0.f, 0.f};
#pragma unroll
  for (int t = 0; t < NT; t++) acc[t] = zero8;

  for (int k0 = 0; k0 < IN_F; k0 += 4) {
    v2f a = *(const v2f*)(Xrow + k0 + ksel);   // A[m][k0+ksel], A[m][k0+ksel+1]
#pragma unroll
    for (int t = 0; t < NT; t++) {
      v2f b = *(const v2f*)(&Wl[(t * 16 + m) * LSTR + k0 + ksel]);  // B[k][n], B[k+1][n]
      acc[t] = __builtin_amdgcn_wmma_f32_16x16x4_f32(
          false, a, false, b, (short)0, acc[t], false, false);
    }
  }

  const int rhalf = (lane >> 4) * 8;           // D rows: rhalf + r
#pragma unroll
  for (int t = 0; t < NT; t++) {
    int col = t * 16 + m;
    float bv = (MODE == 0) ? 0.0f : bias[col];
#pragma unroll
    for (int r = 0; r < 8; r++) {
      int row = rowBase + rhalf + r;
      if (row < nrows) {
        float v = acc[t][r];
        if (MODE == 1) v = fmaxf(v + bv, 0.0f);
        else if (MODE == 2) v = 0.5f * (v + bv);
        Y[(size_t)row * FOUT + col] = v;
      }
    }
  }
}

// ---------------- edge scatter: agg[dst] += h[src] * dis[src]*dis[dst] ----------------
__global__ __launch_bounds__(256) void k_edge_scatter128(
    const float* __restrict__ h, const int* __restrict__ src, const int* __restrict__ dst,
    const float* __restrict__ dis, float* __restrict__ agg, int e) {
  int gw = (int)(((size_t)blockIdx.x * blockDim.x + threadIdx.x) >> 5);
  int lane = threadIdx.x & 31;
  if (gw >= e) return;
  int s = src[gw], d = dst[gw];
  float nrm = dis[s] * dis[d];
  v4f v = *(const v4f*)(h + (size_t)s * 128 + lane * 4);
  float* out = agg + (size_t)d * 128 + lane * 4;
  atomicAdd(out + 0, v.x * nrm);
  atomicAdd(out + 1, v.y * nrm);
  atomicAdd(out + 2, v.z * nrm);
  atomicAdd(out + 3, v.w * nrm);
}

__global__ __launch_bounds__(256) void k_edge_scatter64(
    const float* __restrict__ h, const int* __restrict__ src, const int* __restrict__ dst,
    const float* __restrict__ dis, float* __restrict__ agg, int e) {
  int gw = (int)(((size_t)blockIdx.x * blockDim.x + threadIdx.x) >> 5);
  int lane = threadIdx.x & 31;
  if (gw >= e) return;
  int s = src[gw], d = dst[gw];
  float nrm = dis[s] * dis[d];
  v2f v = *(const v2f*)(h + (size_t)s * 64 + lane * 2);
  float* out = agg + (size_t)d * 64 + lane * 2;
  atomicAdd(out + 0, v.x * nrm);
  atomicAdd(out + 1, v.y * nrm);
}

// a1 = relu(agg1 + h1*di + b1), in place into agg1  (128 features)
__global__ void k_post_relu128(float* __restrict__ a, const float* __restrict__ h1,
                               const float* __restrict__ di, const float* __restrict__ b, int n) {
  long i = (long)blockIdx.x * blockDim.x + threadIdx.x;
  if (i < (long)n * 128) {
    int node = (int)(i >> 7), f = (int)(i & 127);
    a[i] = fmaxf(a[i] + h1[i] * di[node] + b[f], 0.0f);
  }
}

// out += 0.5*(agg2 + h2*di + b2)   (64 features; out already holds 0.5*z_mlp)
__global__ void k_final64(float* __restrict__ out, const float* __restrict__ agg2,
                          const float* __restrict__ h2, const float* __restrict__ di,
                          const float* __restrict__ b2, int n) {
  long i = (long)blockIdx.x * blockDim.x + threadIdx.x;
  if (i < (long)n * 64) {
    int node = (int)(i >> 6), f = (int)(i & 63);
    out[i] += 0.5f * (agg2[i] + h2[i] * di[node] + b2[f]);
  }
}

extern "C" void kernel_launch(void* const* d_in, const int* in_sizes, int n_in,
                              void* d_out, int out_size, void* d_ws, size_t ws_size,
                              hipStream_t stream) {
  const float* x   = (const float*)d_in[0];
  const int*   ei  = (const int*)d_in[1];
  const float* gW1 = (const float*)d_in[2];
  const float* gb1 = (const float*)d_in[3];
  const float* gW2 = (const float*)d_in[4];
  const float* gb2 = (const float*)d_in[5];
  const float* mW1 = (const float*)d_in[6];
  const float* mb1 = (const float*)d_in[7];
  const float* mW2 = (const float*)d_in[8];
  const float* mb2 = (const float*)d_in[9];

  const int N = in_sizes[0] / IN_F;
  const int E = in_sizes[1] / 2;
  const int* src = ei;
  const int* dst = ei + E;

  float* deg  = (float*)d_ws;
  float* dis  = deg + N;
  float* di   = dis + N;
  float* h1   = di + N;                         // N*128
  float* agg1 = h1 + (size_t)N * 128;           // N*128 (becomes a1)
  float* m1   = agg1 + (size_t)N * 128;         // N*128
  float* h2   = h1;                             // reuse (h1 dead after post_relu)
  float* agg2 = h1 + (size_t)N * 64;            // second half of old h1 region

  float* out = (float*)d_out;

  const int T = 256;
  dim3 blk(T);
  int nbN   = (N + T - 1) / T;
  int nbE   = (E + T - 1) / T;
  int nbEd  = (E * 32 + T - 1) / T;             // 1 wave per edge
  int nbG   = (N + 127) / 128;                  // GEMM blocks (128 rows each)
  long n128 = (long)N * 128, n64 = (long)N * 64;
  int nbZ1  = (int)((n128 / 4 + T - 1) / T);
  int nbZ2  = (int)((n64 / 4 + T - 1) / T);
  int nbP1  = (int)((n128 + T - 1) / T);
  int nbF   = (int)((n64 + T - 1) / T);

  // degrees
  k_deg_init<<<nbN, blk, 0, stream>>>(deg, N);
  k_deg_count<<<nbE, blk, 0, stream>>>(dst, deg, E);
  k_deg_fin<<<nbN, blk, 0, stream>>>(deg, dis, di, N);

  // GCN layer 1
  k_gemm_wmma<128, 0><<<nbG, blk, 0, stream>>>(x, gW1, gb1, h1, N);
  k_zero4<<<nbZ1, blk, 0, stream>>>((v4f*)agg1, n128 / 4);
  k_edge_scatter128<<<nbEd, blk, 0, stream>>>(h1, src, dst, dis, agg1, E);
  k_post_relu128<<<nbP1, blk, 0, stream>>>(agg1, h1, di, gb1, N);   // agg1 -> a1

  // GCN layer 2 matmul (h1 region is now free for h2/agg2)
  k_gemm_wmma<64, 0><<<nbG, blk, 0, stream>>>(agg1, gW2, gb2, h2, N);

  // MLP branch: out = 0.5 * (relu(x@mW1+mb1) @ mW2 + mb2)
  k_gemm_wmma<128, 1><<<nbG, blk, 0, stream>>>(x, mW1, mb1, m1, N);
  k_gemm_wmma<64, 2><<<nbG, blk, 0, stream>>>(m1, mW2, mb2, out, N);

  // GCN layer 2 aggregation + fusion
  k_zero4<<<nbZ2, blk, 0, stream>>>((v4f*)agg2, n64 / 4);
  k_edge_scatter64<<<nbEd, blk, 0, stream>>>(h2, src, dst, dis, agg2, E);
  k_final64<<<nbF, blk, 0, stream>>>(out, agg2, h2, di, gb2, N);

  (void)n_in; (void)out_size; (void)ws_size;
}